// Sample_51247549776070
// MI455X (gfx1250) — compile-verified
//
#include <hip/hip_runtime.h>
#include <stdint.h>

// ---------------------------------------------------------------------------
// Problem constants (match the reference).
// ---------------------------------------------------------------------------
#define BATCH     8
#define NPTS      65536
#define NSAMPLE   1024
#define NTHREADS  1024
#define NGROUP    32          // groups per distance sweep; 2 points/thread/group

// ---------------------------------------------------------------------------
// Dynamic LDS layout (total 311,584 B <= 320KB/WGP):
//   [0, 262144)          float dists[NPTS]           (persistent)
//   [262144, 262160)     float q[4]                  (current query point)
//   [262160, 262416)     u64 sred[32]                (per-wave reduction)
//   [262432, 311584)     stage[2][3][1024] 8B slots  (async double buffer)
// ---------------------------------------------------------------------------
#define Q_OFF      (NPTS * 4)
#define RED_OFF    (Q_OFF + 16)
#define STAGE_OFF  (RED_OFF + 272)     // 262432, 16B aligned
#define STAGE_CH   8192                // bytes per channel plane (1024 * 8)
#define STAGE_BUF  (3 * STAGE_CH)      // 24576 bytes per buffer
#define SMEM_BYTES (STAGE_OFF + 2 * STAGE_BUF)

// ---------------------------------------------------------------------------
// CDNA5 async global->LDS copy (ASYNCcnt-tracked). VDST operand is the raw
// per-lane LDS byte address; VADDR is the 64-bit global address.
// ---------------------------------------------------------------------------
__device__ __forceinline__ void async_load_b64(uint32_t lds_addr,
                                               const float* gaddr) {
  asm volatile("global_load_async_to_lds_b64 %0, %1, off"
               :
               : "v"(lds_addr), "v"(gaddr)
               : "memory");
}

__device__ __forceinline__ void wait_async_le3() {
  asm volatile("s_wait_asynccnt 0x3" ::: "memory");
}

// Issue one group's coordinate stream (x,y,z; 2 consecutive points per lane).
__device__ __forceinline__ void issue_group(const float* __restrict__ px,
                                            const float* __restrict__ py,
                                            const float* __restrict__ pz,
                                            uint32_t lane_base, int g, int buf,
                                            int tid) {
  const int n0 = (g << 11) + (tid << 1);               // g*2048 + tid*2
  const uint32_t a = lane_base + (uint32_t)buf * STAGE_BUF;
  async_load_b64(a,                px + n0);
  async_load_b64(a + STAGE_CH,     py + n0);
  async_load_b64(a + 2 * STAGE_CH, pz + n0);
}

// ---------------------------------------------------------------------------
// Packed argmax key: dist >= 0 so float bits order like uint; (~idx) makes
// ties pick the SMALLEST index (jnp.argmax first-occurrence semantics).
// ---------------------------------------------------------------------------
__device__ __forceinline__ unsigned long long pack_key(float d, unsigned n) {
  return ((unsigned long long)__float_as_uint(d) << 32) |
         (unsigned long long)(0xFFFFFFFFu - n);
}

__device__ __forceinline__ unsigned long long max_u64(unsigned long long a,
                                                      unsigned long long b) {
  return a > b ? a : b;
}

// wave32 xor-shuffle max reduction on packed u64 keys (DS permute path).
__device__ __forceinline__ unsigned long long wave_reduce_max(unsigned long long k) {
#pragma unroll
  for (int m = 16; m >= 1; m >>= 1) {
    unsigned lo = (unsigned)k;
    unsigned hi = (unsigned)(k >> 32);
    unsigned olo = (unsigned)__shfl_xor((int)lo, m, 32);
    unsigned ohi = (unsigned)__shfl_xor((int)hi, m, 32);
    unsigned long long o =
        ((unsigned long long)ohi << 32) | (unsigned long long)olo;
    k = max_u64(k, o);
  }
  return k;
}

// ---------------------------------------------------------------------------
// Kernel 1: furthest point sampling, one 1024-thread workgroup per batch.
// dists persistent in LDS; coordinates streamed L2 -> LDS with CDNA5 async
// copies, double buffered, prefetched one group ahead (wrapping across the
// per-step reduction barriers since the stream is iteration-invariant).
// ---------------------------------------------------------------------------
__global__ __launch_bounds__(NTHREADS, 1)
void fps_kernel(const float* __restrict__ pts, int* __restrict__ idx_out) {
  const int b   = blockIdx.x;
  const int tid = threadIdx.x;

  const float* __restrict__ px = pts + (size_t)b * 3 * NPTS;
  const float* __restrict__ py = px + NPTS;
  const float* __restrict__ pz = px + 2 * NPTS;

  extern __shared__ unsigned char smraw[];
  float2* __restrict__ sdist = (float2*)smraw;         // dists as 8B pairs
  float4* __restrict__ sdist4 = (float4*)smraw;        // for fast init
  float* smq = (float*)(smraw + Q_OFF);
  unsigned long long* sred = (unsigned long long*)(smraw + RED_OFF);

  // Raw LDS byte address of this lane's stage slot (for the async VDST field).
  const uint32_t lane_base =
      __builtin_amdgcn_groupstaticsize() + STAGE_OFF + (uint32_t)tid * 8;

  // ---- init persistent dists + seed point (index 0) ----
  const float4 big = make_float4(1e10f, 1e10f, 1e10f, 1e10f);
#pragma unroll
  for (int c = 0; c < 16; ++c) sdist4[c * NTHREADS + tid] = big;
  if (tid == 0) {
    smq[0] = px[0];
    smq[1] = py[0];
    smq[2] = pz[0];
    idx_out[b * NSAMPLE] = 0;
  }
  __syncthreads();

  // Prologue: prefetch group 0 into buffer 0.
  issue_group(px, py, pz, lane_base, 0, 0, tid);

  for (int i = 1; i < NSAMPLE; ++i) {
    const float qx = smq[0], qy = smq[1], qz = smq[2];
    unsigned long long best = 0;

    for (int g = 0; g < NGROUP; ++g) {
      // Prefetch the next group (wraps to group 0 of the NEXT step at g=31,
      // hiding L2 latency across the reduction barriers).
      const int gn = (g + 1) & (NGROUP - 1);
      issue_group(px, py, pz, lane_base, gn, gn & 1, tid);
      wait_async_le3();   // ASYNCcnt in-order: <=3 left => group g has landed

      const unsigned char* sg =
          smraw + STAGE_OFF + (unsigned)(g & 1) * STAGE_BUF + (unsigned)tid * 8;
      const float2 x = *(const float2*)(sg);
      const float2 y = *(const float2*)(sg + STAGE_CH);
      const float2 z = *(const float2*)(sg + 2 * STAGE_CH);

      const int n0 = (g << 11) + (tid << 1);
      const int di = (g << 10) + tid;
      float2 dc = sdist[di];
      {
        float ax = x.x - qx, ay = y.x - qy, az = z.x - qz;
        dc.x = fminf(dc.x, fmaf(ax, ax, fmaf(ay, ay, az * az)));
      }
      {
        float ax = x.y - qx, ay = y.y - qy, az = z.y - qz;
        dc.y = fminf(dc.y, fmaf(ax, ax, fmaf(ay, ay, az * az)));
      }
      sdist[di] = dc;

      best = max_u64(best, max_u64(pack_key(dc.x, (unsigned)n0),
                                   pack_key(dc.y, (unsigned)n0 + 1)));
    }

    // intra-wave reduction (DS permute), then cross-wave via LDS
    best = wave_reduce_max(best);
    if ((tid & 31) == 0) sred[tid >> 5] = best;
    __syncthreads();

    if (tid < 32) {
      unsigned long long k = sred[tid];
      k = wave_reduce_max(k);
      if (tid == 0) {
        const unsigned w = 0xFFFFFFFFu - (unsigned)(k & 0xFFFFFFFFu);
        idx_out[b * NSAMPLE + i] = (int)w;
        smq[0] = px[w];
        smq[1] = py[w];
        smq[2] = pz[w];
      }
    }
    __syncthreads();   // q + dists visible before next step
  }
  // 3 dangling prefetch asyncs are drained by S_ENDPGM's implicit wait-idle.
}

// ---------------------------------------------------------------------------
// Kernel 2: gather selected points: out[b,c,k] = pts[b,c,idx[b,k]]
// ---------------------------------------------------------------------------
__global__ __launch_bounds__(256)
void fps_gather_kernel(const float* __restrict__ pts,
                       const int* __restrict__ idx,
                       float* __restrict__ out) {
  const int i = blockIdx.x * blockDim.x + threadIdx.x;
  if (i >= BATCH * 3 * NSAMPLE) return;
  const int k  = i & (NSAMPLE - 1);
  const int bc = i >> 10;               // b*3 + c
  const int b  = bc / 3;
  const int src = idx[b * NSAMPLE + k];
  out[i] = pts[(size_t)bc * NPTS + src];
}

// ---------------------------------------------------------------------------
extern "C" void kernel_launch(void* const* d_in, const int* in_sizes, int n_in,
                              void* d_out, int out_size, void* d_ws,
                              size_t ws_size, hipStream_t stream) {
  (void)in_sizes; (void)n_in; (void)out_size; (void)ws_size;
  const float* pts = (const float*)d_in[0];   // (B, 3, N) fp32
  float* out = (float*)d_out;                 // (B, 3, NSAMPLE) fp32
  int* idx = (int*)d_ws;                      // 8*1024 int32 scratch (32KB)

  fps_kernel<<<BATCH, NTHREADS, SMEM_BYTES, stream>>>(pts, idx);

  const int total = BATCH * 3 * NSAMPLE;
  fps_gather_kernel<<<(total + 255) / 256, 256, 0, stream>>>(pts, idx, out);
}